// PointPillarsScatter_11458972746018
// MI455X (gfx1250) — compile-verified
//
#include <hip/hip_runtime.h>

// PointPillarsScatter for MI455X (gfx1250, wave32).
// Pure bandwidth op: no FLOPs -> no WMMA. Strategy = inverse-map gather so the
// 256MB output is written once, coalesced, with nontemporal b128 stores, and
// the 41MB feature table stays resident in the 192MB L2 (read from HBM once).

typedef float f4v __attribute__((ext_vector_type(4)));
typedef int   i4v __attribute__((ext_vector_type(4)));

#define BATCH 4
#define NX    512
#define NY    512
#define CH    64
#define PLANE (NX * NY)   // 262144 elements per (b,c) plane

// ---------------- gather path (preferred) ----------------

// Fill inverse map with -1 (empty pillar). 16B stores.
__global__ void pps_fill_map(i4v* __restrict__ map4, int n4) {
    int i = blockIdx.x * blockDim.x + threadIdx.x;
    if (i < n4) {
        i4v m = { -1, -1, -1, -1 };
        map4[i] = m;
    }
}

// map[b*PLANE + x*NY + y] = voxel_id   (coords row = [z, x, y, b])
__global__ void pps_scatter_ids(const i4v* __restrict__ coords,
                                int* __restrict__ map, int nv) {
    int v = blockIdx.x * blockDim.x + threadIdx.x;
    if (v >= nv) return;
    i4v c = coords[v];                 // .x=z .y=x .z=y .w=b
    map[c.w * PLANE + c.y * NY + c.z] = v;
}

// One thread = 4 channels x 4 y-positions of output (16 floats).
// 1x int4 map load, 4x float4 gathers from L2-resident features,
// 4x4 register transpose, 4x float4 nontemporal streaming stores.
__global__ void pps_gather(const float* __restrict__ vf,
                           const int* __restrict__ map,
                           float* __restrict__ out) {
    unsigned t  = blockIdx.x * blockDim.x + threadIdx.x;
    unsigned y0 = (t & 127u) << 2;          // 128 y-groups of 4
    unsigned x  = (t >> 7) & 511u;          // 512 x
    unsigned cg = (t >> 16) & 15u;          // 16 channel-groups of 4
    unsigned b  = t >> 20;                  // 4 batches

    i4v m = *(const i4v*)(map + b * PLANE + x * NY + y0);
    const f4v* vf4 = (const f4v*)vf;        // feature row = 16 f4v (64 floats)

    f4v zero = 0.0f;
    f4v g0 = (m.x >= 0) ? vf4[(unsigned)m.x * 16u + cg] : zero;
    f4v g1 = (m.y >= 0) ? vf4[(unsigned)m.y * 16u + cg] : zero;
    f4v g2 = (m.z >= 0) ? vf4[(unsigned)m.z * 16u + cg] : zero;
    f4v g3 = (m.w >= 0) ? vf4[(unsigned)m.w * 16u + cg] : zero;

    size_t base = (size_t)(b * CH + cg * 4u) * PLANE + (size_t)x * NY + y0;
    f4v r0 = { g0.x, g1.x, g2.x, g3.x };
    f4v r1 = { g0.y, g1.y, g2.y, g3.y };
    f4v r2 = { g0.z, g1.z, g2.z, g3.z };
    f4v r3 = { g0.w, g1.w, g2.w, g3.w };
    __builtin_nontemporal_store(r0, (f4v*)(out + base));
    __builtin_nontemporal_store(r1, (f4v*)(out + base + (size_t)PLANE));
    __builtin_nontemporal_store(r2, (f4v*)(out + base + (size_t)2 * PLANE));
    __builtin_nontemporal_store(r3, (f4v*)(out + base + (size_t)3 * PLANE));
}

// ---------------- fallback direct path (if ws too small) ----------------

__global__ void pps_zero_out(f4v* __restrict__ out4, int n4) {
    int i = blockIdx.x * blockDim.x + threadIdx.x;
    if (i < n4) out4[i] = (f4v)0.0f;
}

// 16 threads per voxel; each handles 4 channels (one float4 of the row).
__global__ void pps_scatter_direct(const f4v* __restrict__ vf4,
                                   const i4v* __restrict__ coords,
                                   float* __restrict__ out, int nv) {
    int t = blockIdx.x * blockDim.x + threadIdx.x;
    int v = t >> 4;
    if (v >= nv) return;
    int cg = t & 15;
    i4v c  = coords[v];                    // z, x, y, b
    f4v f  = vf4[(unsigned)v * 16u + cg];
    size_t base = (size_t)(c.w * CH + cg * 4) * PLANE + (size_t)c.y * NY + c.z;
    out[base]                     = f.x;
    out[base + (size_t)PLANE]     = f.y;
    out[base + (size_t)2 * PLANE] = f.z;
    out[base + (size_t)3 * PLANE] = f.w;
}

extern "C" void kernel_launch(void* const* d_in, const int* in_sizes, int n_in,
                              void* d_out, int out_size, void* d_ws, size_t ws_size,
                              hipStream_t stream) {
    const float* vf     = (const float*)d_in[0];   // [NV, 64] f32
    const int*   coords = (const int*)d_in[1];     // [NV, 4]  i32
    float*       out    = (float*)d_out;           // [4, 64, 512, 512] f32

    int nv = in_sizes[0] / CH;                     // 160000

    size_t map_bytes = (size_t)BATCH * PLANE * sizeof(int);   // 4 MB
    if (ws_size >= map_bytes) {
        int* map = (int*)d_ws;
        int n4 = (BATCH * PLANE) / 4;                          // 262144
        pps_fill_map  <<<(n4 + 255) / 256, 256, 0, stream>>>((i4v*)map, n4);
        pps_scatter_ids<<<(nv + 255) / 256, 256, 0, stream>>>((const i4v*)coords, map, nv);
        unsigned total = (unsigned)BATCH * (CH / 4) * NX * (NY / 4); // 4,194,304
        pps_gather    <<<total / 256, 256, 0, stream>>>(vf, map, out);
    } else {
        // Fallback: zero canvas then direct transposed scatter.
        int n4 = out_size / 4;                                 // 16,777,216
        pps_zero_out      <<<(n4 + 255) / 256, 256, 0, stream>>>((f4v*)out, n4);
        pps_scatter_direct<<<(nv * 16 + 255) / 256, 256, 0, stream>>>(
            (const f4v*)vf, (const i4v*)coords, out, nv);
    }
}